// GCN_60163901882953
// MI455X (gfx1250) — compile-verified
//
#include <hip/hip_runtime.h>
#include <hip/hip_bf16.h>

// ---------------------------------------------------------------------------
// GCN (3x GCNConv -> concat -> Linear+ReLU -> Linear -> softmax) for MI455X.
//
// Precision: fp32 throughout (matches reference); dense GEMMs run on the
// CDNA5 tensor path via V_WMMA_F32_16X16X4_F32.
// Memory: scatter phase is the roofline limiter (~2.6 GB of random f32
// atomic traffic, L2-resident since h is 76.8 MB < 192 MB L2).
// ---------------------------------------------------------------------------

typedef __attribute__((ext_vector_type(2))) float v2f;
typedef __attribute__((ext_vector_type(8))) float v8f;

#define NN 50000
#define EE 800000
#define HD 128
#define CD 384

// ---------------- degree / normalization ----------------
__global__ void deg_init(float* deg, int n) {
    int i = blockIdx.x * blockDim.x + threadIdx.x;
    if (i < n) deg[i] = 1.0f;                       // self-loop contributes 1
}
__global__ void deg_count(const int* __restrict__ dst, float* deg, int e) {
    int i = blockIdx.x * blockDim.x + threadIdx.x;
    if (i < e) atomicAdd(&deg[dst[i]], 1.0f);
}
__global__ void deg_rsqrt(float* deg, int n) {
    int i = blockIdx.x * blockDim.x + threadIdx.x;
    if (i < n) deg[i] = rsqrtf(deg[i]);             // deg >= 1 always
}

// ---------------- fp32 WMMA GEMM: C[MxNc] = A[MxK] * B ----------------
// One wave computes one 16x16 tile. Tile counts are exact (N=3125*16), so
// EXEC is all-ones at every WMMA (ISA requirement).
// A-fragment layout (ISA 7.12.2, 32-bit A 16x4): lane&15 = M row,
//   VGPR0 = K+{0,2}, VGPR1 = K+{1,3} across the two lane-halves ->
//   per lane: two consecutive K at k0 + 2*(lane>>4).
// B-fragment mirrors it: lane&15 = N col, halves pick K rows.
// C/D: VGPR i holds row (i + 8*(lane>>4)), col = lane&15.
template <bool BT>
__global__ void __launch_bounds__(256)
gemm16_f32(const float* __restrict__ A, int lda,
           const float* __restrict__ B, int ldb,   // BT: B is [Nc x K], ldb=K
           float* __restrict__ C, int ldc,
           int K, int tilesPerRow, int totalTiles) {
    int wave = threadIdx.x >> 5;
    int lane = threadIdx.x & 31;
    int tile = blockIdx.x * (blockDim.x >> 5) + wave;
    if (tile >= totalTiles) return;                 // wave-uniform exit
    int tm = (tile / tilesPerRow) << 4;
    int tn = (tile % tilesPerRow) << 4;
    int rc   = lane & 15;                           // A row / B-C col
    int half = lane >> 4;

    v8f acc = {};
    const float* arow = A + (size_t)(tm + rc) * lda;
#pragma unroll 4
    for (int k0 = 0; k0 < K; k0 += 4) {
        int ka = k0 + 2 * half;
        v2f a;
        a.x = arow[ka];
        a.y = arow[ka + 1];
        v2f b;
        if (BT) {
            const float* bp = B + (size_t)(tn + rc) * ldb + ka;
            b.x = bp[0];
            b.y = bp[1];
        } else {
            b.x = B[(size_t)(ka + 0) * ldb + tn + rc];
            b.y = B[(size_t)(ka + 1) * ldb + tn + rc];
        }
        acc = __builtin_amdgcn_wmma_f32_16x16x4_f32(
            false, a, false, b, (short)0, acc, false, false);
    }
#pragma unroll
    for (int i = 0; i < 8; ++i)
        C[(size_t)(tm + i + 8 * half) * ldc + tn + rc] = acc[i];
}

// ---------------- self-loop + bias init: h[v] = dinv[v]^2 * xw[v] + b ------
__global__ void self_init(const float* __restrict__ XW, const float* __restrict__ dinv,
                          const float* __restrict__ bias,
                          float* __restrict__ H, int ldh, int colOff, int n) {
    int idx = blockIdx.x * blockDim.x + threadIdx.x;
    int v = idx >> 5, lane = idx & 31;              // 32 lanes x float4 = 128
    if (v >= n) return;
    float dn = dinv[v];
    float nn = dn * dn;
    float4 x = ((const float4*)(XW + (size_t)v * HD))[lane];
    float4 b = ((const float4*)bias)[lane];
    float* op = H + (size_t)v * ldh + colOff + lane * 4;
    op[0] = x.x * nn + b.x;
    op[1] = x.y * nn + b.y;
    op[2] = x.z * nn + b.z;
    op[3] = x.w * nn + b.w;
}

// ---------------- edge scatter: h[dst] += norm * xw[src] -------------------
__global__ void __launch_bounds__(256)
edge_scatter(const float* __restrict__ XW, const float* __restrict__ dinv,
             const int* __restrict__ src, const int* __restrict__ dst,
             float* __restrict__ H, int ldh, int colOff, int e) {
    int idx = blockIdx.x * blockDim.x + threadIdx.x;
    int eid = idx >> 5, lane = idx & 31;
    if (eid >= e) return;
    int s = src[eid], d = dst[eid];
    float norm = dinv[s] * dinv[d];
    float4 v = ((const float4*)(XW + (size_t)s * HD))[lane];
    float* op = H + (size_t)d * ldh + colOff + lane * 4;
    atomicAdd(op + 0, v.x * norm);
    atomicAdd(op + 1, v.y * norm);
    atomicAdd(op + 2, v.z * norm);
    atomicAdd(op + 3, v.w * norm);
}

// ---------------- logits init with lb2 ----------------
__global__ void logits_init(float* logits, const float* __restrict__ lb2, int n) {
    int i = blockIdx.x * blockDim.x + threadIdx.x;
    if (i < n) {
        logits[2 * i + 0] = lb2[0];
        logits[2 * i + 1] = lb2[1];
    }
}

// ------- fused: t = relu(h @ lW1^T + lb1); logits += t @ lW2^T -------------
// Same WMMA tile structure (K=384, transposed-B access into lW1). Epilogue
// reduces each output row's 16-column slice against lW2 via 16-lane shuffles,
// then one atomicAdd per (row, class) per tile.
__global__ void __launch_bounds__(256)
linear_fused(const float* __restrict__ H,     // [N x 384]
             const float* __restrict__ lW1,   // [384 x 384] (out, in)
             const float* __restrict__ lb1,   // [384]
             const float* __restrict__ lW2,   // [2 x 384]
             float* __restrict__ logits,      // [N x 2], pre-filled with lb2
             int totalTiles) {
    const int K = CD, tilesPerRow = CD / 16;
    int wave = threadIdx.x >> 5;
    int lane = threadIdx.x & 31;
    int tile = blockIdx.x * (blockDim.x >> 5) + wave;
    if (tile >= totalTiles) return;
    int tm = (tile / tilesPerRow) << 4;
    int tn = (tile % tilesPerRow) << 4;
    int col  = lane & 15;
    int half = lane >> 4;

    v8f acc = {};
    const float* arow = H + (size_t)(tm + col) * K;     // col doubles as A-row
    const float* brow = lW1 + (size_t)(tn + col) * K;   // transposed-B access
#pragma unroll 4
    for (int k0 = 0; k0 < K; k0 += 4) {
        int ka = k0 + 2 * half;
        v2f a;
        a.x = arow[ka];
        a.y = arow[ka + 1];
        v2f b;
        b.x = brow[ka];
        b.y = brow[ka + 1];
        acc = __builtin_amdgcn_wmma_f32_16x16x4_f32(
            false, a, false, b, (short)0, acc, false, false);
    }

    float bias = lb1[tn + col];
    float w0 = lW2[0 * CD + tn + col];
    float w1 = lW2[1 * CD + tn + col];
#pragma unroll
    for (int i = 0; i < 8; ++i) {
        float t = acc[i] + bias;
        t = t > 0.0f ? t : 0.0f;
        float p0 = t * w0;
        float p1 = t * w1;
#pragma unroll
        for (int m = 8; m >= 1; m >>= 1) {          // reduce within 16-lane half
            p0 += __shfl_xor(p0, m, 16);
            p1 += __shfl_xor(p1, m, 16);
        }
        if (col == 0) {
            int r = tm + i + 8 * half;
            atomicAdd(&logits[2 * r + 0], p0);
            atomicAdd(&logits[2 * r + 1], p1);
        }
    }
}

// ---------------- softmax over 2 classes ----------------
__global__ void softmax2(const float* __restrict__ logits, float* __restrict__ probs, int n) {
    int i = blockIdx.x * blockDim.x + threadIdx.x;
    if (i >= n) return;
    float a = logits[2 * i + 0], b = logits[2 * i + 1];
    float m = fmaxf(a, b);
    float ea = __expf(a - m), eb = __expf(b - m);
    float inv = 1.0f / (ea + eb);
    probs[2 * i + 0] = ea * inv;
    probs[2 * i + 1] = eb * inv;
}

extern "C" void kernel_launch(void* const* d_in, const int* in_sizes, int n_in,
                              void* d_out, int out_size, void* d_ws, size_t ws_size,
                              hipStream_t stream) {
    const float* x   = (const float*)d_in[0];
    const int*   ei  = (const int*)d_in[1];
    const float* W1  = (const float*)d_in[2];
    const float* b1  = (const float*)d_in[3];
    const float* W2  = (const float*)d_in[4];
    const float* b2  = (const float*)d_in[5];
    const float* W3  = (const float*)d_in[6];
    const float* b3  = (const float*)d_in[7];
    const float* lW1 = (const float*)d_in[8];
    const float* lb1 = (const float*)d_in[9];
    const float* lW2 = (const float*)d_in[10];
    const float* lb2 = (const float*)d_in[11];

    const int N = NN, E = EE;
    const int* src = ei;
    const int* dst = ei + E;

    // workspace: dinv [N] | xw [N*128] | h [N*384]  (~103 MB total)
    float* dinv = (float*)d_ws;
    float* xw   = dinv + N;
    float* h    = xw + (size_t)N * HD;

    float* logits = (float*)d_out;                  // [N,2]
    float* probs  = logits + (size_t)N * 2;         // [N,2]

    const int T = 256;
    const int rowTiles = N / 16;                    // 3125 exactly

    // degrees + dinv
    deg_init<<<(N + T - 1) / T, T, 0, stream>>>(dinv, N);
    deg_count<<<(E + T - 1) / T, T, 0, stream>>>(dst, dinv, E);
    deg_rsqrt<<<(N + T - 1) / T, T, 0, stream>>>(dinv, N);

    // --- conv 1: x[64] -> h1 (h cols 0..127) ---
    {
        int tiles = rowTiles * (HD / 16);
        gemm16_f32<false><<<(tiles + 7) / 8, T, 0, stream>>>(
            x, 64, W1, HD, xw, HD, 64, HD / 16, tiles);
        self_init<<<((size_t)N * 32 + T - 1) / T, T, 0, stream>>>(xw, dinv, b1, h, CD, 0, N);
        edge_scatter<<<((size_t)E * 32 + T - 1) / T, T, 0, stream>>>(
            xw, dinv, src, dst, h, CD, 0, E);
    }
    // --- conv 2: h1 -> h2 (cols 128..255) ---
    {
        int tiles = rowTiles * (HD / 16);
        gemm16_f32<false><<<(tiles + 7) / 8, T, 0, stream>>>(
            h + 0, CD, W2, HD, xw, HD, HD, HD / 16, tiles);
        self_init<<<((size_t)N * 32 + T - 1) / T, T, 0, stream>>>(xw, dinv, b2, h, CD, HD, N);
        edge_scatter<<<((size_t)E * 32 + T - 1) / T, T, 0, stream>>>(
            xw, dinv, src, dst, h, CD, HD, E);
    }
    // --- conv 3: h2 -> h3 (cols 256..383) ---
    {
        int tiles = rowTiles * (HD / 16);
        gemm16_f32<false><<<(tiles + 7) / 8, T, 0, stream>>>(
            h + HD, CD, W3, HD, xw, HD, HD, HD / 16, tiles);
        self_init<<<((size_t)N * 32 + T - 1) / T, T, 0, stream>>>(xw, dinv, b3, h, CD, 2 * HD, N);
        edge_scatter<<<((size_t)E * 32 + T - 1) / T, T, 0, stream>>>(
            xw, dinv, src, dst, h, CD, 2 * HD, E);
    }

    // --- MLP head: relu(h @ lW1^T + lb1) @ lW2^T + lb2, fused ---
    logits_init<<<(N + T - 1) / T, T, 0, stream>>>(logits, lb2, N);
    {
        int tiles = rowTiles * (CD / 16);           // 75000
        linear_fused<<<(tiles + 7) / 8, T, 0, stream>>>(h, lW1, lb1, lW2, logits, tiles);
    }
    softmax2<<<(N + T - 1) / T, T, 0, stream>>>(logits, probs, N);
}